// ContrastLoss_89893665505512
// MI455X (gfx1250) — compile-verified
//
#include <hip/hip_runtime.h>
#include <hip/hip_bf16.h>

// ---------------------------------------------------------------------------
// ContrastLoss for MI455X (gfx1250).
//
// Phase 1: G = X * X^T via V_WMMA_F32_16X16X4_F32 (full-f32 WMMA).
//          X = feature_contrast viewed as [672, 512]  (672 = 32*21).
// Phase 2: closed-form per-step loss from prefix sums over G (no scan),
//          faithfully reproducing the reference's naive exp/log numerics
//          (including inf/inf -> nan and nan*0 -> nan propagation).
// ---------------------------------------------------------------------------

typedef float v2f __attribute__((ext_vector_type(2)));
typedef float v8f __attribute__((ext_vector_type(8)));

#define NROW 672   // 32 * 21
#define DDIM 512
#define NC   20
#define NB   32

// One wave per 16x16 output tile of G. 42x42 tiles cover 672x672 exactly.
__global__ void __launch_bounds__(32)
gram_wmma_kernel(const float* __restrict__ X, float* __restrict__ G) {
    const int m0   = blockIdx.x * 16;
    const int n0   = blockIdx.y * 16;
    const int lane = threadIdx.x;          // 0..31 (wave32)
    const int r    = lane & 15;            // row-in-tile for A, col-in-tile for B
    const int kb   = (lane >> 4) * 2;      // lanes 0-15: K={0,1}; lanes 16-31: K={2,3}

    const float* __restrict__ arow = X + (size_t)(m0 + r) * DDIM + kb;
    const float* __restrict__ brow = X + (size_t)(n0 + r) * DDIM + kb;

    v8f acc = {};
#pragma unroll 8
    for (int k0 = 0; k0 < DDIM; k0 += 4) {
        v2f a = *(const v2f*)(arow + k0);   // A[m, k0+kb .. k0+kb+1]
        v2f b = *(const v2f*)(brow + k0);   // B[k0+kb .. , n] == X[n, k0+kb ..]
        // D = A(16x4,f32) x B(4x16,f32) + C ; args: neg_a,A,neg_b,B,c_mod,C,reuse_a,reuse_b
        acc = __builtin_amdgcn_wmma_f32_16x16x4_f32(
            false, a, false, b, (short)0, acc, false, false);
    }

    // C/D layout: VGPR v -> row = v + 8*(lane/16), col = lane%16
    const int col     = n0 + (lane & 15);
    const int rowbase = m0 + (lane >> 4) * 8;
#pragma unroll
    for (int v = 0; v < 8; ++v) {
        G[(size_t)(rowbase + v) * NROW + col] = acc[v];
    }
}

// One thread per (i,c) step; steps are independent given G.
__global__ void __launch_bounds__(640)
loss_kernel(const float* __restrict__ G, const int* __restrict__ lab,
            float* __restrict__ out) {
    __shared__ int   act[NB][NC];
    __shared__ float npos_inv[NB];
    __shared__ float red[NB * NC];

    const int t = threadIdx.x;                 // 0..639
    act[t / NC][t % NC] = lab[t];
    __syncthreads();
    if (t < NB) {
        int s = 0;
#pragma unroll
        for (int c = 0; c < NC; ++c) s += act[t][c];
        npos_inv[t] = 1.0f / fmaxf((float)s, 1.0f);
    }
    __syncthreads();

    const int i = t / NC;                      // batch index of this step
    const int c = t % NC;                      // class index of this step
    const int col = i * 21 + c + 1;            // Gram column of current fg feature

    float pos = 0.0f;
    float negsum = 0.0f;
    for (int cp = 0; cp < NC; ++cp) {
        int   cnt = 0;
        float afg = 0.0f, abg = 0.0f;
        for (int ip = 0; ip < i; ++ip) {       // committed history (batches < i)
            if (act[ip][cp]) {
                ++cnt;
                afg += G[(size_t)(ip * 21 + cp + 1) * NROW + col];
                abg += G[(size_t)(ip * 21)          * NROW + col];
            }
        }
        // current batch: classes cp<c already committed; cp==c is the tentative
        // slot-0 write which happens even for inactive steps (rolled back later,
        // but the loss is computed from the tentative state).
        const int inc = (cp < c) ? act[i][cp] : ((cp == c) ? 1 : 0);
        if (inc) {
            ++cnt;
            afg += G[(size_t)(i * 21 + cp + 1) * NROW + col];
            abg += G[(size_t)(i * 21)          * NROW + col];
        }
        if (cnt > 0) {
            const float rc  = 1.0f / (float)cnt;
            const float dfg = afg * rc;
            const float dbg = abg * rc;
            if (cp == c) pos = dfg;            // row bi excluded from negatives
            else         negsum += expf(dfg);  // fg negative
            negsum += expf(dbg);               // bg rows (incl. 2c) are negatives
        }
    }

    const float ep  = expf(pos);               // overflows to inf for small cnt,
    const float div = ep + negsum;             // exactly like the reference
    const float la  = -logf(ep / div);         // inf/inf -> nan, as in reference
    const float inv = act[i][c] ? npos_inv[i] : 0.0f;
    red[t] = la * inv;                         // nan * 0 -> nan (matches jnp)
    __syncthreads();

    if (t == 0) {                              // accumulate in scan order
        float loss = 0.0f;
        for (int s = 0; s < NB * NC; ++s) loss += red[s];
        out[0] = loss / (float)NB;
    }
}

extern "C" void kernel_launch(void* const* d_in, const int* in_sizes, int n_in,
                              void* d_out, int out_size, void* d_ws, size_t ws_size,
                              hipStream_t stream) {
    const float* X   = (const float*)d_in[0];   // [32, 21, 512] f32
    const int*   lab = (const int*)d_in[1];     // [32, 20] i32
    float*       G   = (float*)d_ws;            // 672*672 f32 = ~1.8 MB
    float*       out = (float*)d_out;           // scalar loss

    (void)in_sizes; (void)n_in; (void)out_size; (void)ws_size;

    dim3 grid(NROW / 16, NROW / 16);            // 42 x 42 tiles
    gram_wmma_kernel<<<grid, 32, 0, stream>>>(X, G);
    loss_kernel<<<1, NB * NC, 0, stream>>>(G, lab, out);
}